// WaveFieldAttention_85899345920398
// MI455X (gfx1250) — compile-verified
//
#include <hip/hip_runtime.h>
#include <math.h>

// ---------------------------------------------------------------------------
// DeltaCorrection forward for MI455X (gfx1250, wave32, WMMA f32 16x16x4).
// B=4 H=16 N=4096 d=64 C=64  -> 64 bh sequences x 64 chunks.
//   K1: per-chunk gates + (beta*knorm) + scores/mask/intra + kv (4096 blocks)
//   K2: sequential per-(b,h) decay scan, stores pre-chunk S     (  64 blocks)
//   K3: inter = q @ S^T, out = (intra + inter) * out_scale      (4096 blocks)
// All WMMA A/B fragments are contiguous ds_load_b64; one A-fragment is shared
// by the wave's two output tiles (1.5 DS ops per WMMA).
// ---------------------------------------------------------------------------

typedef __attribute__((ext_vector_type(2))) float v2f;
typedef __attribute__((ext_vector_type(8))) float v8f;

#define PITCH 68              // 64 + 4 floats: conflict-free strided access
#define CHUNK_ELEMS 4096      // 64 tokens * 64 dims

__device__ __forceinline__ v8f wmma4(v2f a, v2f b, v8f c) {
  // D = A(16x4 f32) * B(4x16 f32) + C(16x16 f32); exact fp32 accumulate.
  return __builtin_amdgcn_wmma_f32_16x16x4_f32(
      /*neg_a=*/false, a, /*neg_b=*/false, b,
      /*c_mod=*/(short)0, c, /*reuse_a=*/false, /*reuse_b=*/false);
}

// ---------------------------------------------------------------------------
// K1: one block per (bh, chunk). 256 threads = 8 waves. Wave w owns output
// tiles (tm = w>>1, tn = 2*(w&1)) and (tm, tn+1): the pair shares A.
// ---------------------------------------------------------------------------
__global__ __launch_bounds__(256)
void dc_chunk_kernel(const float* __restrict__ q, const float* __restrict__ k,
                     const float* __restrict__ v,
                     const float* __restrict__ w_decay, const float* __restrict__ b_decay,
                     const float* __restrict__ w_write, const float* __restrict__ b_write,
                     float* __restrict__ out_intra,   // d_out, holds unscaled intra
                     float* __restrict__ kv_g,        // ws: per-chunk (d x d) kv
                     float* __restrict__ dec_g) {     // ws: per-chunk decay scalar
  extern __shared__ float lds[];
  float* q_s    = lds;                  // 64 x PITCH  q chunk, row-major
  float* kbn_s  = q_s   + 64 * PITCH;   // 64 x PITCH  kbn_s[c][dd] = beta[c]*knorm[c][dd]
  float* kbT_s  = kbn_s + 64 * PITCH;   // 64 x PITCH  kbT_s[dd][c] = beta[c]*knorm[c][dd]
  float* vT_s   = kbT_s + 64 * PITCH;   // 64 x PITCH  vT_s[dd][c]  = v[c][dd]
  float* s_s    = vT_s  + 64 * PITCH;   // 64 x PITCH  raw k staging, then masked scores
  float* beta_s = s_s   + 64 * PITCH;   // 64
  float* alph_s = beta_s + 64;          // 64
  float* invn_s = alph_s + 64;          // 64
  float* wd_s   = invn_s + 64;          // 64
  float* ww_s   = wd_s   + 64;          // 64

  const int tid  = threadIdx.x;
  const size_t base = (size_t)blockIdx.x * CHUNK_ELEMS;

  const int r  = tid >> 2;              // token/row 0..63
  const int cb = (tid & 3) * 16;        // column base 0/16/32/48

  // ---- Phase A0: load q (row-major), k (staged), v (transposed directly)
  {
    const float4* q4 = (const float4*)(q + base + r * 64 + cb);
    const float4* k4 = (const float4*)(k + base + r * 64 + cb);
    const float4* v4 = (const float4*)(v + base + r * 64 + cb);
    float4* qs = (float4*)(q_s + r * PITCH + cb);
    float4* ks = (float4*)(s_s + r * PITCH + cb);
#pragma unroll
    for (int i = 0; i < 4; ++i) {
      qs[i] = q4[i];
      ks[i] = k4[i];
      const float4 vv = v4[i];
      vT_s[(cb + 4 * i + 0) * PITCH + r] = vv.x;   // transpose v on the fly
      vT_s[(cb + 4 * i + 1) * PITCH + r] = vv.y;
      vT_s[(cb + 4 * i + 2) * PITCH + r] = vv.z;
      vT_s[(cb + 4 * i + 3) * PITCH + r] = vv.w;
    }
    if (tid < 64) { wd_s[tid] = w_decay[tid]; ww_s[tid] = w_write[tid]; }
  }
  __syncthreads();

  // ---- Phase A1: per-token gates + norm (one thread per token)
  if (tid < 64) {
    float dw = 0.f, ww = 0.f, ss = 0.f;
    const float4* kr = (const float4*)(s_s + tid * PITCH);
#pragma unroll
    for (int i = 0; i < 16; ++i) {
      const float4 kk = kr[i];
      const float4 wd = ((const float4*)wd_s)[i];
      const float4 wr = ((const float4*)ww_s)[i];
      dw += kk.x * wd.x + kk.y * wd.y + kk.z * wd.z + kk.w * wd.w;
      ww += kk.x * wr.x + kk.y * wr.y + kk.z * wr.z + kk.w * wr.w;
      ss += kk.x * kk.x + kk.y * kk.y + kk.z * kk.z + kk.w * kk.w;
    }
    beta_s[tid] = 1.f / (1.f + __expf(-(ww + b_write[0])));
    alph_s[tid] = 1.f / (1.f + __expf(-(dw + b_decay[0])));
    invn_s[tid] = 1.f / fmaxf(sqrtf(ss), 1e-12f);
  }
  __syncthreads();

  // ---- Phase A2: build beta*knorm in both orientations; thread 0: decay
  {
    const float sc = beta_s[r] * invn_s[r];
    const float4* kr = (const float4*)(s_s + r * PITCH + cb);
    float4* kn = (float4*)(kbn_s + r * PITCH + cb);
#pragma unroll
    for (int i = 0; i < 4; ++i) {
      float4 kk = kr[i];
      kk.x *= sc; kk.y *= sc; kk.z *= sc; kk.w *= sc;
      kn[i] = kk;
      kbT_s[(cb + 4 * i + 0) * PITCH + r] = kk.x;
      kbT_s[(cb + 4 * i + 1) * PITCH + r] = kk.y;
      kbT_s[(cb + 4 * i + 2) * PITCH + r] = kk.z;
      kbT_s[(cb + 4 * i + 3) * PITCH + r] = kk.w;
    }
  }
  if (tid == 0) {
    float m = 0.f;
    for (int i = 0; i < 64; ++i) m += alph_s[i];
    m *= (1.f / 64.f);
#pragma unroll
    for (int i = 0; i < 6; ++i) m = m * m;     // mean^64
    dec_g[blockIdx.x] = m;
  }
  __syncthreads();

  const int lane = tid & 31;
  const int wv   = tid >> 5;
  const int l15  = lane & 15;
  const int lh   = lane >> 4;                  // 0: low half, 1: high half
  const int tm   = wv >> 1;
  const int row  = tm * 16 + l15;              // shared A row for both tiles
  const int col0 = (wv & 1) * 32 + l15;        // tile tn and tn+1
  const int col1 = col0 + 16;

  // ---- Step 1: scores = q @ (beta*knorm)^T, causal mask, into s_s
  {
    v8f a0 = {}, a1 = {};
    for (int k0 = 0; k0 < 64; k0 += 4) {
      const int kb = k0 + 2 * lh;
      v2f av = *(const v2f*)(q_s   + row  * PITCH + kb);
      v2f b0 = *(const v2f*)(kbn_s + col0 * PITCH + kb);
      v2f b1 = *(const v2f*)(kbn_s + col1 * PITCH + kb);
      a0 = wmma4(av, b0, a0);
      a1 = wmma4(av, b1, a1);
    }
#pragma unroll
    for (int e = 0; e < 8; ++e) {
      const int i = tm * 16 + lh * 8 + e;
      s_s[i * PITCH + col0] = (col0 <= i) ? a0[e] : 0.f;
      s_s[i * PITCH + col1] = (col1 <= i) ? a1[e] : 0.f;
    }
  }
  __syncthreads();

  // ---- Step 2a: intra = scores @ v  -> global (unscaled)
  {
    v8f a0 = {}, a1 = {};
    for (int k0 = 0; k0 < 64; k0 += 4) {
      const int kb = k0 + 2 * lh;
      v2f av = *(const v2f*)(s_s  + row  * PITCH + kb);
      v2f b0 = *(const v2f*)(vT_s + col0 * PITCH + kb);
      v2f b1 = *(const v2f*)(vT_s + col1 * PITCH + kb);
      a0 = wmma4(av, b0, a0);
      a1 = wmma4(av, b1, a1);
    }
#pragma unroll
    for (int e = 0; e < 8; ++e) {
      const int i = tm * 16 + lh * 8 + e;
      out_intra[base + i * 64 + col0] = a0[e];
      out_intra[base + i * 64 + col1] = a1[e];
    }
  }

  // ---- Step 2b: kv[i][j] = sum_c v[c][i] * (beta[c]*knorm[c][j]) -> ws
  {
    v8f a0 = {}, a1 = {};
    for (int k0 = 0; k0 < 64; k0 += 4) {
      const int kc = k0 + 2 * lh;
      v2f av = *(const v2f*)(vT_s  + row  * PITCH + kc);   // A[i][c] = v[c][i]
      v2f b0 = *(const v2f*)(kbT_s + col0 * PITCH + kc);   // B[c][j] = beta*knorm
      v2f b1 = *(const v2f*)(kbT_s + col1 * PITCH + kc);
      a0 = wmma4(av, b0, a0);
      a1 = wmma4(av, b1, a1);
    }
#pragma unroll
    for (int e = 0; e < 8; ++e) {
      const int i = tm * 16 + lh * 8 + e;
      kv_g[base + i * 64 + col0] = a0[e];
      kv_g[base + i * 64 + col1] = a1[e];
    }
  }
}

// ---------------------------------------------------------------------------
// K2: sequential scan per (b,h). Stores PRE-update S for every chunk.
// ---------------------------------------------------------------------------
__global__ __launch_bounds__(256)
void dc_scan_kernel(const float* __restrict__ kv_g, const float* __restrict__ dec_g,
                    float* __restrict__ S_g) {
  const int bh  = blockIdx.x;
  const int tid = threadIdx.x;
  const int off = tid * 16;            // 256 threads x 16 floats = 4096
  float4 S[4] = {};
  for (int n = 0; n < 64; ++n) {
    const size_t gb = ((size_t)bh * 64 + n) * CHUNK_ELEMS + off;
    const float dec = dec_g[bh * 64 + n];
#pragma unroll
    for (int i = 0; i < 4; ++i) {
      ((float4*)(S_g + gb))[i] = S[i];                     // state BEFORE chunk n
      const float4 kv = ((const float4*)(kv_g + gb))[i];
      S[i].x = dec * S[i].x + kv.x;
      S[i].y = dec * S[i].y + kv.y;
      S[i].z = dec * S[i].z + kv.z;
      S[i].w = dec * S[i].w + kv.w;
    }
  }
}

// ---------------------------------------------------------------------------
// K3: inter[c][i] = sum_j S[i][j] q[c][j]; out = (intra + inter) * scale
// ---------------------------------------------------------------------------
__global__ __launch_bounds__(256)
void dc_inter_kernel(const float* __restrict__ q, const float* __restrict__ S_g,
                     const float* __restrict__ out_scale, float* __restrict__ out) {
  extern __shared__ float lds[];
  float* q_s = lds;                 // 64 x PITCH
  float* S_s = lds + 64 * PITCH;    // 64 x PITCH, row-major S[i][j]
  const int tid = threadIdx.x;
  const size_t base = (size_t)blockIdx.x * CHUNK_ELEMS;
  {
    const int r  = tid >> 2;
    const int cb = (tid & 3) * 16;
#pragma unroll
    for (int i = 0; i < 4; ++i) {
      ((float4*)(q_s + r * PITCH + cb))[i] = ((const float4*)(q   + base + r * 64 + cb))[i];
      ((float4*)(S_s + r * PITCH + cb))[i] = ((const float4*)(S_g + base + r * 64 + cb))[i];
    }
  }
  __syncthreads();

  const float sc = out_scale[0];
  const int lane = tid & 31, wv = tid >> 5;
  const int l15 = lane & 15, lh = lane >> 4;
  const int tm   = wv >> 1;
  const int row  = tm * 16 + l15;            // c (token); A shared by tile pair
  const int col0 = (wv & 1) * 32 + l15;      // i (feature)
  const int col1 = col0 + 16;

  v8f a0 = {}, a1 = {};
  for (int k0 = 0; k0 < 64; k0 += 4) {
    const int kb = k0 + 2 * lh;
    v2f av = *(const v2f*)(q_s + row  * PITCH + kb);   // q[c][j]
    v2f b0 = *(const v2f*)(S_s + col0 * PITCH + kb);   // B[k=j][n=i] = S[i][j]
    v2f b1 = *(const v2f*)(S_s + col1 * PITCH + kb);
    a0 = wmma4(av, b0, a0);
    a1 = wmma4(av, b1, a1);
  }
#pragma unroll
  for (int e = 0; e < 8; ++e) {
    const int c = tm * 16 + lh * 8 + e;
    const size_t o0 = base + c * 64 + col0;
    const size_t o1 = base + c * 64 + col1;
    out[o0] = (out[o0] + a0[e]) * sc;
    out[o1] = (out[o1] + a1[e]) * sc;
  }
}

// ---------------------------------------------------------------------------
extern "C" void kernel_launch(void* const* d_in, const int* in_sizes, int n_in,
                              void* d_out, int out_size, void* d_ws, size_t ws_size,
                              hipStream_t stream) {
  const float* q         = (const float*)d_in[0];
  const float* k         = (const float*)d_in[1];
  const float* v         = (const float*)d_in[2];
  const float* w_decay   = (const float*)d_in[3];
  const float* b_decay   = (const float*)d_in[4];
  const float* w_write   = (const float*)d_in[5];
  const float* b_write   = (const float*)d_in[6];
  const float* out_scale = (const float*)d_in[7];
  float* out = (float*)d_out;

  const size_t per = (size_t)64 * 64 * CHUNK_ELEMS;   // 16,777,216 floats
  float* kv_g  = (float*)d_ws;        // 64 MB
  float* S_g   = kv_g + per;          // 64 MB
  float* dec_g = S_g  + per;          // 16 KB

  const size_t lds1 = (size_t)(5 * 64 * PITCH + 5 * 64) * sizeof(float);  // ~88.3 KB
  const size_t lds3 = (size_t)(2 * 64 * PITCH) * sizeof(float);           // ~34.8 KB

  dc_chunk_kernel<<<4096, 256, lds1, stream>>>(q, k, v, w_decay, b_decay,
                                               w_write, b_write, out, kv_g, dec_g);
  dc_scan_kernel<<<64, 256, 0, stream>>>(kv_g, dec_g, S_g);
  dc_inter_kernel<<<4096, 256, lds3, stream>>>(q, S_g, out_scale, out);
}